// ResNet_63866163691992
// MI455X (gfx1250) — compile-verified
//
#include <hip/hip_runtime.h>

typedef __attribute__((ext_vector_type(2))) float v2f;
typedef __attribute__((ext_vector_type(8))) float v8f;

#define BLOCK 128
#define NWAVES 4

__device__ __forceinline__ v8f zero8() {
  v8f z = {0.f, 0.f, 0.f, 0.f, 0.f, 0.f, 0.f, 0.f};
  return z;
}

__device__ __forceinline__ v8f wmma4(v2f a, v2f b, v8f c) {
  // D = A(16x4 f32) * B(4x16 f32) + C(16x16 f32)
  return __builtin_amdgcn_wmma_f32_16x16x4_f32(false, a, false, b, (short)0, c,
                                               false, false);
}

__device__ __forceinline__ float siluf(float x) {
  return x * (1.0f / (1.0f + __expf(-x)));
}

// LDS pool layout (floats):
//   P  @ 0     [16][48]   raw p per event (i*12+l)
//   X  @ 768   [16][384]  td output, layout l*32+c  (aliased by T [16][192])
//   D0 @ 6912  [16][192]  pre-activation d0, layout j*32+c  (aliased by V0)
//   D  @ 9984  [16][192]  current silu(d)                   (aliased by V)
//   Q0 @ 13056 [16][96]
//   Q  @ 14592 [16][96]
#define LDS_FLOATS 16128

__global__ __launch_bounds__(BLOCK) void resnet_fused(
    const float* __restrict__ p, const float* __restrict__ da,
    const float* __restrict__ qa, const float* __restrict__ va,
    const float* __restrict__ Wtd, const float* __restrict__ btd,
    const float* __restrict__ Wdb, const float* __restrict__ bdb,
    const float* __restrict__ Wdr1, const float* __restrict__ bdr1,
    const float* __restrict__ Wdr2, const float* __restrict__ bdr2,
    const float* __restrict__ Wtq, const float* __restrict__ btq,
    const float* __restrict__ Wqb, const float* __restrict__ bqb,
    const float* __restrict__ Wqr1, const float* __restrict__ bqr1,
    const float* __restrict__ Wqr2, const float* __restrict__ bqr2,
    const float* __restrict__ Wtv, const float* __restrict__ btv,
    const float* __restrict__ Wv1, const float* __restrict__ bv1,
    const float* __restrict__ Wv2, const float* __restrict__ bv2,
    const float* __restrict__ Wvs, const float* __restrict__ bvs,
    float* __restrict__ out, int N) {
  __shared__ float lds[LDS_FLOATS];
  float* P = lds;          // [16][48]
  float* X = lds + 768;    // [16][384]
  float* D0 = lds + 6912;  // [16][192]
  float* D = lds + 9984;   // [16][192]
  float* Q0 = lds + 13056; // [16][96]
  float* Q = lds + 14592;  // [16][96]
  float* T = X;            // [16][192] tq output (p1 dead by then)
  float* V = D;            // [16][192] head (d dead by then)
  float* V0 = D0;          // [16][192]

  const int tid = threadIdx.x;
  const int lane = tid & 31;
  // wave id is wave-uniform: force it into an SGPR so all unit loops use
  // scalar branches and EXEC stays all-1s around every WMMA.
  const int wid = __builtin_amdgcn_readfirstlane(tid >> 5);
  const int e16 = lane & 15; // WMMA column = event in tile
  const int hl = lane >> 4;  // half-wave select (K pair / M+8)
  const int n0 = blockIdx.x * 16;
  int gev = n0 + e16;
  if (gev >= N) gev = N - 1;

  // ---- stage raw p into LDS (variant transforms applied at read time) ----
  for (int i = tid; i < 16 * 48; i += BLOCK) {
    int e = i / 48, f = i - e * 48;
    int g = n0 + e;
    if (g >= N) g = N - 1;
    P[i] = p[g * 48 + f];
  }

  // per-wave register accumulators for symmetrized silu(q) (up to 2 units)
  v8f qs0 = zero8(), qs1 = zero8();

  for (int var = 0; var < 4; ++var) {
    const float s1 = (var >= 2) ? -1.f : 1.f;             // eta flip (ch 1)
    const float s2 = (var == 1 || var == 2) ? -1.f : 1.f; // phi flip (ch 2)
    __syncthreads();

    // ================= td: X[32,12] = Wtd[32,4] @ p[4,12] + b ============
    for (int u = wid; u < 24; u += NWAVES) {
      int l = u >> 1, mb = (u & 1) << 4;
      int kA = hl << 1; // K pair base (channels kA, kA+1)
      v2f a, b;
      a.x = Wtd[(mb + e16) * 4 + kA];
      a.y = Wtd[(mb + e16) * 4 + kA + 1];
      float b0 = P[e16 * 48 + kA * 12 + l];
      float b1 = P[e16 * 48 + (kA + 1) * 12 + l];
      // branchless symmetry transform:
      //   lanes with kA==0 hold channels (0,1): ch1 *= s1
      //   lanes with kA==2 hold channels (2,3): ch2 -> rotate(+0) * s2
      float rot = b0 + 1.f;
      rot = rot - 2.f * floorf(rot * 0.5f);
      rot = (rot - 1.f) * s2;
      b.x = (kA == 2) ? rot : b0;
      b.y = (kA == 0) ? (b1 * s1) : b1;
      v8f acc = zero8();
      acc = wmma4(a, b, acc);
      int cb = mb + (hl << 3);
      int base = e16 * 384 + l * 32 + cb;
#pragma unroll
      for (int jj = 0; jj < 8; ++jj) X[base + jj] = acc[jj] + btd[cb + jj];
    }
    __syncthreads();

    // ================= db: d0 = Wdb[30,64] @ X(pairs), stride 2 ==========
    for (int u = wid; u < 12; u += NWAVES) {
      int j = u >> 1, mb = (u & 1) << 4;
      int row = mb + e16;
      int rc = row < 30 ? row : 29;
      float msk = row < 30 ? 1.f : 0.f;
      v8f acc = zero8();
      for (int t = 0; t < 16; ++t) {
        int kA = (t << 2) + (hl << 1); // kk = i*2+k, kA even
        v2f a, b;
        a.x = Wdb[rc * 64 + kA] * msk;
        a.y = Wdb[rc * 64 + kA + 1] * msk;
        int i = kA >> 1;
        b.x = X[e16 * 384 + (2 * j) * 32 + i];
        b.y = X[e16 * 384 + (2 * j + 1) * 32 + i];
        acc = wmma4(a, b, acc);
      }
      int base = e16 * 192 + j * 32 + mb + (hl << 3);
#pragma unroll
      for (int jj = 0; jj < 8; ++jj) D0[base + jj] = acc[jj];
    }
    __syncthreads();
    // elementwise: bias on rows<30, da on rows 30/31; D = silu(D0)
    for (int idx = tid; idx < 16 * 192; idx += BLOCK) {
      int e = idx / 192, f = idx - e * 192;
      int j = f >> 5, c = f & 31;
      int g = n0 + e;
      if (g >= N) g = N - 1;
      float val = (c < 30) ? (D0[idx] + bdb[c]) : da[g * 12 + (c - 30) * 6 + j];
      D0[idx] = val;
      D[idx] = siluf(val);
    }
    __syncthreads();

    // ================= dijet residual blocks (K = 32ch x 3 taps) =========
    auto dr_stage = [&](const float* __restrict__ W,
                        const float* __restrict__ bb) {
      v8f r0 = zero8(), r1 = zero8(), r2 = zero8();
      int cnt = 0;
      for (int u = wid; u < 12; u += NWAVES, ++cnt) {
        int j = u >> 1, mb = (u & 1) << 4;
        v8f acc = zero8();
        for (int t = 0; t < 24; ++t) {
          int kA = (t << 2) + (hl << 1);
          v2f a, b;
          a.x = W[(mb + e16) * 96 + kA];
          a.y = W[(mb + e16) * 96 + kA + 1];
          int i0 = kA / 3, k0 = kA - 3 * i0;
          int kB = kA + 1;
          int i1 = kB / 3, k1 = kB - 3 * i1;
          b.x = (k0 < 2) ? X[e16 * 384 + (2 * j + k0) * 32 + i0]
                         : D[e16 * 192 + j * 32 + i0];
          b.y = (k1 < 2) ? X[e16 * 384 + (2 * j + k1) * 32 + i1]
                         : D[e16 * 192 + j * 32 + i1];
          acc = wmma4(a, b, acc);
        }
        int cb = mb + (hl << 3);
        int base = e16 * 192 + j * 32 + cb;
#pragma unroll
        for (int jj = 0; jj < 8; ++jj) acc[jj] += bb[cb + jj] + D0[base + jj];
        if (cnt == 0) r0 = acc;
        else if (cnt == 1) r1 = acc;
        else r2 = acc;
      }
      __syncthreads(); // all reads of D done before rewriting it
      cnt = 0;
      for (int u = wid; u < 12; u += NWAVES, ++cnt) {
        int j = u >> 1, mb = (u & 1) << 4;
        v8f acc = (cnt == 0) ? r0 : ((cnt == 1) ? r1 : r2);
        int base = e16 * 192 + j * 32 + mb + (hl << 3);
#pragma unroll
        for (int jj = 0; jj < 8; ++jj) D[base + jj] = siluf(acc[jj]);
      }
      __syncthreads();
    };
    dr_stage(Wdr1, bdr1);
    dr_stage(Wdr2, bdr2);

    // ================= tq: T[32,6] = Wtq[32,32] @ D + b ==================
    for (int u = wid; u < 12; u += NWAVES) {
      int j = u >> 1, mb = (u & 1) << 4;
      v8f acc = zero8();
      for (int t = 0; t < 8; ++t) {
        int kA = (t << 2) + (hl << 1);
        v2f a, b;
        a.x = Wtq[(mb + e16) * 32 + kA];
        a.y = Wtq[(mb + e16) * 32 + kA + 1];
        b.x = D[e16 * 192 + j * 32 + kA];
        b.y = D[e16 * 192 + j * 32 + kA + 1];
        acc = wmma4(a, b, acc);
      }
      int cb = mb + (hl << 3);
      int base = e16 * 192 + j * 32 + cb;
#pragma unroll
      for (int jj = 0; jj < 8; ++jj) T[base + jj] = acc[jj] + btq[cb + jj];
    }
    __syncthreads();

    // ================= qb: q0 = Wqb[30,64] @ T(pairs), stride 2 ==========
    for (int u = wid; u < 6; u += NWAVES) {
      int j = u >> 1, mb = (u & 1) << 4;
      int row = mb + e16;
      int rc = row < 30 ? row : 29;
      float msk = row < 30 ? 1.f : 0.f;
      v8f acc = zero8();
      for (int t = 0; t < 16; ++t) {
        int kA = (t << 2) + (hl << 1);
        v2f a, b;
        a.x = Wqb[rc * 64 + kA] * msk;
        a.y = Wqb[rc * 64 + kA + 1] * msk;
        int i = kA >> 1;
        b.x = T[e16 * 192 + (2 * j) * 32 + i];
        b.y = T[e16 * 192 + (2 * j + 1) * 32 + i];
        acc = wmma4(a, b, acc);
      }
      int base = e16 * 96 + j * 32 + mb + (hl << 3);
#pragma unroll
      for (int jj = 0; jj < 8; ++jj) Q0[base + jj] = acc[jj];
    }
    __syncthreads();
    for (int idx = tid; idx < 16 * 96; idx += BLOCK) {
      int e = idx / 96, f = idx - e * 96;
      int j = f >> 5, c = f & 31;
      int g = n0 + e;
      if (g >= N) g = N - 1;
      float val = (c < 30) ? (Q0[idx] + bqb[c]) : qa[g * 6 + (c - 30) * 3 + j];
      Q0[idx] = val;
      Q[idx] = siluf(val);
    }
    __syncthreads();

    // ================= qr1 (residual, K = 32 x 3) ========================
    {
      v8f r0 = zero8(), r1 = zero8();
      int cnt = 0;
      for (int u = wid; u < 6; u += NWAVES, ++cnt) {
        int j = u >> 1, mb = (u & 1) << 4;
        v8f acc = zero8();
        for (int t = 0; t < 24; ++t) {
          int kA = (t << 2) + (hl << 1);
          v2f a, b;
          a.x = Wqr1[(mb + e16) * 96 + kA];
          a.y = Wqr1[(mb + e16) * 96 + kA + 1];
          int i0 = kA / 3, k0 = kA - 3 * i0;
          int kB = kA + 1;
          int i1 = kB / 3, k1 = kB - 3 * i1;
          b.x = (k0 < 2) ? T[e16 * 192 + (2 * j + k0) * 32 + i0]
                         : Q[e16 * 96 + j * 32 + i0];
          b.y = (k1 < 2) ? T[e16 * 192 + (2 * j + k1) * 32 + i1]
                         : Q[e16 * 96 + j * 32 + i1];
          acc = wmma4(a, b, acc);
        }
        int cb = mb + (hl << 3);
        int base = e16 * 96 + j * 32 + cb;
#pragma unroll
        for (int jj = 0; jj < 8; ++jj) acc[jj] += bqr1[cb + jj] + Q0[base + jj];
        if (cnt == 0) r0 = acc;
        else r1 = acc;
      }
      __syncthreads();
      cnt = 0;
      for (int u = wid; u < 6; u += NWAVES, ++cnt) {
        int j = u >> 1, mb = (u & 1) << 4;
        v8f acc = (cnt == 0) ? r0 : r1;
        int base = e16 * 96 + j * 32 + mb + (hl << 3);
#pragma unroll
        for (int jj = 0; jj < 8; ++jj) Q[base + jj] = siluf(acc[jj]);
      }
      __syncthreads();
    }

    // ================= qr2 -> accumulate silu(q) in registers ============
    {
      int cnt = 0;
      for (int u = wid; u < 6; u += NWAVES, ++cnt) {
        int j = u >> 1, mb = (u & 1) << 4;
        v8f acc = zero8();
        for (int t = 0; t < 24; ++t) {
          int kA = (t << 2) + (hl << 1);
          v2f a, b;
          a.x = Wqr2[(mb + e16) * 96 + kA];
          a.y = Wqr2[(mb + e16) * 96 + kA + 1];
          int i0 = kA / 3, k0 = kA - 3 * i0;
          int kB = kA + 1;
          int i1 = kB / 3, k1 = kB - 3 * i1;
          b.x = (k0 < 2) ? T[e16 * 192 + (2 * j + k0) * 32 + i0]
                         : Q[e16 * 96 + j * 32 + i0];
          b.y = (k1 < 2) ? T[e16 * 192 + (2 * j + k1) * 32 + i1]
                         : Q[e16 * 96 + j * 32 + i1];
          acc = wmma4(a, b, acc);
        }
        int cb = mb + (hl << 3);
        int base = e16 * 96 + j * 32 + cb;
#pragma unroll
        for (int jj = 0; jj < 8; ++jj) {
          float val = siluf(acc[jj] + bqr2[cb + jj] + Q0[base + jj]);
          if (cnt == 0) qs0[jj] += val;
          else qs1[jj] += val;
        }
      }
    }
  } // variants

  // ---- write symmetrized average q into Q ----
  __syncthreads();
  {
    int cnt = 0;
    for (int u = wid; u < 6; u += NWAVES, ++cnt) {
      int j = u >> 1, mb = (u & 1) << 4;
      int base = e16 * 96 + j * 32 + mb + (hl << 3);
#pragma unroll
      for (int jj = 0; jj < 8; ++jj)
        Q[base + jj] = ((cnt == 0) ? qs0[jj] : qs1[jj]) * 0.25f;
    }
  }
  __syncthreads();

  // ================= head tv: V[64,3] = Wtv[64,34] @ concat(q, va3) ======
  for (int u = wid; u < 12; u += NWAVES) {
    int l = u >> 2, mb = (u & 3) << 4;
    v8f acc = zero8();
    for (int t = 0; t < 9; ++t) { // K padded 34 -> 36
      int kA = (t << 2) + (hl << 1);
      int k0 = kA, k1 = kA + 1;
      float m0 = (k0 < 34) ? 1.f : 0.f, m1 = (k1 < 34) ? 1.f : 0.f;
      int k0c = (k0 < 34) ? k0 : 0, k1c = (k1 < 34) ? k1 : 0;
      int row = mb + e16;
      v2f a, b;
      a.x = Wtv[row * 34 + k0c] * m0;
      a.y = Wtv[row * 34 + k1c] * m1;
      b.x = (k0 < 32) ? Q[e16 * 96 + l * 32 + k0]
                      : ((k0 < 34) ? va[gev * 2 + (k0 - 32)] : 0.f);
      b.y = (k1 < 32) ? Q[e16 * 96 + l * 32 + k1]
                      : ((k1 < 34) ? va[gev * 2 + (k1 - 32)] : 0.f);
      acc = wmma4(a, b, acc);
    }
    int cb = mb + (hl << 3);
    int base = e16 * 192 + l * 64 + cb;
#pragma unroll
    for (int jj = 0; jj < 8; ++jj) {
      float r = acc[jj] + btv[cb + jj];
      V0[base + jj] = r;
      V[base + jj] = siluf(r);
    }
  }
  __syncthreads();

  // ================= head residual 1x1 convs (K = 64) ====================
  auto v_stage = [&](const float* __restrict__ W, const float* __restrict__ bb) {
    v8f r0 = zero8(), r1 = zero8(), r2 = zero8();
    int cnt = 0;
    for (int u = wid; u < 12; u += NWAVES, ++cnt) {
      int l = u >> 2, mb = (u & 3) << 4;
      v8f acc = zero8();
      for (int t = 0; t < 16; ++t) {
        int kA = (t << 2) + (hl << 1);
        v2f a, b;
        a.x = W[(mb + e16) * 64 + kA];
        a.y = W[(mb + e16) * 64 + kA + 1];
        b.x = V[e16 * 192 + l * 64 + kA];
        b.y = V[e16 * 192 + l * 64 + kA + 1];
        acc = wmma4(a, b, acc);
      }
      int cb = mb + (hl << 3);
      int base = e16 * 192 + l * 64 + cb;
#pragma unroll
      for (int jj = 0; jj < 8; ++jj) acc[jj] += bb[cb + jj] + V0[base + jj];
      if (cnt == 0) r0 = acc;
      else if (cnt == 1) r1 = acc;
      else r2 = acc;
    }
    __syncthreads();
    cnt = 0;
    for (int u = wid; u < 12; u += NWAVES, ++cnt) {
      int l = u >> 2, mb = (u & 3) << 4;
      v8f acc = (cnt == 0) ? r0 : ((cnt == 1) ? r1 : r2);
      int base = e16 * 192 + l * 64 + mb + (hl << 3);
#pragma unroll
      for (int jj = 0; jj < 8; ++jj) V[base + jj] = siluf(acc[jj]);
    }
    __syncthreads();
  };
  v_stage(Wv1, bv1);
  v_stage(Wv2, bv2);

  // ================= vs: scalar = sum_{c,k} Wvs[c,k] * V[c,k] ============
  {
    int e = tid >> 3, part = tid & 7; // 8 partial lanes per event
    float sum = 0.f;
#pragma unroll
    for (int f = 0; f < 24; ++f) {
      int ff = part * 24 + f;
      int c = ff / 3, k = ff - 3 * c;
      sum += Wvs[ff] * V[e * 192 + k * 64 + c];
    }
    sum += __shfl_xor(sum, 4, 8);
    sum += __shfl_xor(sum, 2, 8);
    sum += __shfl_xor(sum, 1, 8);
    if (part == 0 && (n0 + e) < N) out[n0 + e] = sum + bvs[0];
  }
}

extern "C" void kernel_launch(void* const* d_in, const int* in_sizes, int n_in,
                              void* d_out, int out_size, void* d_ws,
                              size_t ws_size, hipStream_t stream) {
  // setup_inputs order: x(0,unused), p(1), da(2), qa(3), va(4), then weights.
  const float* p = (const float*)d_in[1];
  const float* da = (const float*)d_in[2];
  const float* qa = (const float*)d_in[3];
  const float* va = (const float*)d_in[4];
  const float* Wtd = (const float*)d_in[5];
  const float* btd = (const float*)d_in[6];
  const float* Wdb = (const float*)d_in[7];
  const float* bdb = (const float*)d_in[8];
  const float* Wdr1 = (const float*)d_in[9];
  const float* bdr1 = (const float*)d_in[10];
  const float* Wdr2 = (const float*)d_in[11];
  const float* bdr2 = (const float*)d_in[12];
  const float* Wtq = (const float*)d_in[13];
  const float* btq = (const float*)d_in[14];
  const float* Wqb = (const float*)d_in[15];
  const float* bqb = (const float*)d_in[16];
  const float* Wqr1 = (const float*)d_in[17];
  const float* bqr1 = (const float*)d_in[18];
  const float* Wqr2 = (const float*)d_in[19];
  const float* bqr2 = (const float*)d_in[20];
  const float* Wtv = (const float*)d_in[21];
  const float* btv = (const float*)d_in[22];
  const float* Wv1 = (const float*)d_in[23];
  const float* bv1 = (const float*)d_in[24];
  const float* Wv2 = (const float*)d_in[25];
  const float* bv2 = (const float*)d_in[26];
  const float* Wvs = (const float*)d_in[27];
  const float* bvs = (const float*)d_in[28];

  const int N = in_sizes[1] / 48; // p is [N,4,12]
  const int nblocks = (N + 15) / 16;
  resnet_fused<<<dim3(nblocks), dim3(BLOCK), 0, stream>>>(
      p, da, qa, va, Wtd, btd, Wdb, bdb, Wdr1, bdr1, Wdr2, bdr2, Wtq, btq, Wqb,
      bqb, Wqr1, bqr1, Wqr2, bqr2, Wtv, btv, Wv1, bv1, Wv2, bv2, Wvs, bvs,
      (float*)d_out, N);
}